// GNNLayer_27754078667622
// MI455X (gfx1250) — compile-verified
//
#include <hip/hip_runtime.h>
#include <math.h>

typedef __attribute__((ext_vector_type(2))) float v2f;
typedef __attribute__((ext_vector_type(8))) float v8f;

// ---------------------------------------------------------------------------
// C[M x (NT*16)] = A[M x 128] * B[128 x (NT*16)], fp32 via V_WMMA_F32_16X16X4_F32
// One wave32 owns a 16-row strip of C across all NT column tiles.
//
// B is staged once per workgroup into LDS, pre-swizzled into the WMMA lane
// layout: element (k, n) -> bs[((k>>1)*N + n)*2 + (k&1)], so each lane's
// B operand for one WMMA is a single aligned ds_load_b64.
//
// A (16x4 f32) per-lane layout: lane L (L<16) holds row M=L, K = k0+{0,1};
//                               lane L (L>=16) holds row M=L-16, K = k0+{2,3}.
// C/D (16x16 f32): VGPR v -> M = v + 8*(L>>4), N = L&15.
// ---------------------------------------------------------------------------
template <int NT>
__global__ __launch_bounds__(256)
void gemm128_f32_wmma(const float* __restrict__ A, const float* __restrict__ B,
                      float* __restrict__ C, int M) {
  const int N = NT * 16;                 // 64 or 128 (power of two)
  __shared__ float bs[128 * NT * 16];    // 32 KB (NT=4) / 64 KB (NT=8)

  // ---- stage + swizzle B into LDS (coalesced global reads) ----
  for (int idx = threadIdx.x; idx < 128 * N; idx += 256) {
    int k = idx / N;                     // N is a compile-time power of two
    int n = idx & (N - 1);
    bs[(((k >> 1) * N + n) << 1) + (k & 1)] = B[idx];
  }
  __syncthreads();

  int wave    = (int)((blockIdx.x * blockDim.x + threadIdx.x) >> 5);
  int lane    = (int)(threadIdx.x & 31);
  int rowBase = wave * 16;
  if (rowBase >= M) return;              // after the barrier

  int m  = lane & 15;   // column within tile / row within A tile
  int kh = lane >> 4;   // which K-half this lane-group holds
  int arow = rowBase + m;
  if (arow >= M) arow = M - 1;           // clamp pad rows; stores predicated

  v8f acc[NT] = {};

  for (int k0 = 0; k0 < 128; k0 += 4) {
    const float* ap = A + (size_t)arow * 128 + (k0 + kh * 2);
    v2f a;
    a.x = ap[0];
    a.y = ap[1];
    int pbase = ((k0 >> 1) + kh) * N + m;   // pair row in swizzled LDS image
#pragma unroll
    for (int j = 0; j < NT; ++j) {
      v2f b = *(const v2f*)(bs + ((pbase + j * 16) << 1));  // ds_load_b64
      acc[j] = __builtin_amdgcn_wmma_f32_16x16x4_f32(
          false, a, false, b, (short)0, acc[j], false, false);
    }
  }

  int rOff = kh * 8;
#pragma unroll
  for (int j = 0; j < NT; ++j) {
#pragma unroll
    for (int v = 0; v < 8; ++v) {
      int r = rowBase + rOff + v;
      if (r < M) C[(size_t)r * N + j * 16 + m] = acc[j][v];
    }
  }
}

// ---------------------------------------------------------------------------
// Zero-fill (graph-capture safe, float4 stores)
// ---------------------------------------------------------------------------
__global__ void zero_f32(float* __restrict__ p, size_t n4) {
  size_t i = (size_t)blockIdx.x * blockDim.x + threadIdx.x;
  if (i < n4) {
    float4 z = make_float4(0.f, 0.f, 0.f, 0.f);
    ((float4*)p)[i] = z;
  }
}

// ---------------------------------------------------------------------------
// Edge phase: one wave32 per edge.
//   alpha = sigmoid( relu(nodeAttn[sub] + relAttn[rel]) . w_alpha + b )
//   agg[obj] += alpha * (hidden[sub] + rela_embed[rel])     (float atomics)
// 64 attn dims -> 2/lane; 128 message dims -> 4/lane.
// Entire working set (~80 MB) is resident in the 192 MB L2.
// ---------------------------------------------------------------------------
__global__ __launch_bounds__(256)
void edge_scatter(const float* __restrict__ hidden,
                  const int*   __restrict__ edges,   // [nEdge][6]: 2=rel,4=sub,5=obj
                  const float* __restrict__ rela,
                  const float* __restrict__ nodeAttn,
                  const float* __restrict__ relAttn,
                  const float* __restrict__ w_alpha,
                  const float* __restrict__ w_alpha_b,
                  float* __restrict__ agg,
                  int nEdge) {
  int wave = (int)((blockIdx.x * blockDim.x + threadIdx.x) >> 5);
  int lane = (int)(threadIdx.x & 31);
  if (wave >= nEdge) return;

  const int* e = edges + (size_t)wave * 6;
  int rel = e[2];
  int sub = e[4];
  int obj = e[5];

  // attention score
  float2 an = *(const float2*)(nodeAttn + (size_t)sub * 64 + lane * 2);
  float2 ar = *(const float2*)(relAttn  + (size_t)rel * 64 + lane * 2);
  float h0 = fmaxf(an.x + ar.x, 0.f);
  float h1 = fmaxf(an.y + ar.y, 0.f);
  float2 w = *(const float2*)(w_alpha + lane * 2);
  float part = h0 * w.x + h1 * w.y;
#pragma unroll
  for (int off = 16; off > 0; off >>= 1)
    part += __shfl_xor(part, off, 32);
  float alpha = 1.f / (1.f + __expf(-(part + w_alpha_b[0])));

  // weighted message scatter-add
  float4 hs = *(const float4*)(hidden + (size_t)sub * 128 + lane * 4);
  float4 hr = *(const float4*)(rela   + (size_t)rel * 128 + lane * 4);
  float* dst = agg + (size_t)obj * 128 + lane * 4;
  atomicAdd(dst + 0, alpha * (hs.x + hr.x));
  atomicAdd(dst + 1, alpha * (hs.y + hr.y));
  atomicAdd(dst + 2, alpha * (hs.z + hr.z));
  atomicAdd(dst + 3, alpha * (hs.w + hr.w));
}

// ---------------------------------------------------------------------------
// Launcher
// inputs: 0=hidden(fp32 N*128) 1=edges(int N_E*6) 2=n_node 3=rela_embed(fp32)
//         4=Ws(128x64) 5=Wr(128x64) 6=w_alpha_w(64x1) 7=w_alpha_b(1) 8=Wh(128x128)
// ---------------------------------------------------------------------------
extern "C" void kernel_launch(void* const* d_in, const int* in_sizes, int n_in,
                              void* d_out, int out_size, void* d_ws, size_t ws_size,
                              hipStream_t stream) {
  const float* hidden = (const float*)d_in[0];
  const int*   edges  = (const int*)  d_in[1];
  const float* rela   = (const float*)d_in[3];
  const float* Ws     = (const float*)d_in[4];
  const float* Wr     = (const float*)d_in[5];
  const float* w_aw   = (const float*)d_in[6];
  const float* w_ab   = (const float*)d_in[7];
  const float* Wh     = (const float*)d_in[8];

  const int nNode = in_sizes[0] / 128;
  const int nEdge = in_sizes[1] / 6;
  const int nRel  = in_sizes[3] / 128;
  const int nRelPad = (nRel + 15) & ~15;

  // workspace partition
  float* nodeAttn = (float*)d_ws;                         // nNode   * 64
  float* relAttn  = nodeAttn + (size_t)nNode * 64;        // nRelPad * 64
  float* agg      = relAttn  + (size_t)nRelPad * 64;      // nNode   * 128

  // 1) zero the aggregation buffer
  {
    size_t n4 = ((size_t)nNode * 128) / 4;
    zero_f32<<<(unsigned)((n4 + 255) / 256), 256, 0, stream>>>(agg, n4);
  }

  // 2) nodeAttn = hidden @ Ws   (M=nNode, N=64)
  {
    int strips = (nNode + 15) / 16;
    gemm128_f32_wmma<4><<<(strips + 7) / 8, 256, 0, stream>>>(hidden, Ws, nodeAttn, nNode);
  }

  // 3) relAttn = rela_embed @ Wr   (M=nRel, N=64)
  {
    int strips = (nRel + 15) / 16;
    gemm128_f32_wmma<4><<<(strips + 7) / 8, 256, 0, stream>>>(rela, Wr, relAttn, nRel);
  }

  // 4) per-edge alpha + scatter-add into agg
  {
    int blocks = (nEdge + 7) / 8;   // 8 waves (edges) per 256-thread block
    edge_scatter<<<blocks, 256, 0, stream>>>(hidden, edges, rela, nodeAttn, relAttn,
                                             w_aw, w_ab, agg, nEdge);
  }

  // 5) hidden_new = agg @ Wh   (M=nNode, N=128)
  {
    int strips = (nNode + 15) / 16;
    gemm128_f32_wmma<8><<<(strips + 7) / 8, 256, 0, stream>>>(agg, Wh, (float*)d_out, nNode);
  }
}